// l2_lg_lce_59854664237538
// MI455X (gfx1250) — compile-verified
//
#include <hip/hip_runtime.h>
#include <hip/hip_bf16.h>
#include <math.h>

// ---------------- geometry constants ----------------
#define IMGS      128
#define HH        128
#define WW        512
#define PLANE     65536        // H*W
#define IMGSTRIDE 196608       // 3*PLANE
#define NPIX      8388608      // IMGS*PLANE
#define NGRP      2097152      // NPIX/4 (4-pixel groups)
#define ROWS      4096
#define COLS      10000
#define COLS4     2500

// workspace layout (in floats)
#define OFF_THRES 0            // 256
#define OFF_GS_O  256          // 1024
#define OFF_GS_T  1280         // 1024
#define OFF_L2    2304         // 1024
#define OFF_LG    3328         // 2048
#define OFF_CE    5376         // 4096
#define OFF_END   9472
#define GRAY_BYTE_OFF (OFF_END * 4)             // 37888, 4-byte aligned
#define WS_NEED   (GRAY_BYTE_OFF + 2ull * NPIX) // gray u8 cache for both image sets

#define NBA 1024   // kernel A blocks (8 slices per image)
#define NBB 2048   // kernel B blocks
#define TPB 256

typedef float v2f __attribute__((ext_vector_type(2)));
typedef float v8f __attribute__((ext_vector_type(8)));

// ---------------- DPP xor-add (no LDS, pure VALU) ----------------
// ctrl: quad_perm(1,0,3,2)=0xB1 -> xor1 ; quad_perm(2,3,0,1)=0x4E -> xor2
//       row_xmask:4 = 0x164 -> xor4     ; row_xmask:8 = 0x168 -> xor8
template <int CTRL>
__device__ __forceinline__ float dpp_xadd(float v) {
    int y = __builtin_amdgcn_update_dpp(0, __float_as_int(v), CTRL, 0xF, 0xF, true);
    return v + __int_as_float(y);
}

// ------------- wave32 reduction via V_WMMA_F32_16X16X4_F32 -------------
// A = ones(16x4), B = lane partials (B vgpr1 = 0). D[m][n] = colsum(n), so
// every lane's C[0] = partial[lane%16] + partial[lane%16+16]. The remaining
// 16-column sum is four DPP xor-adds (all within a 16-lane DPP row).
__device__ __forceinline__ float wave_reduce(float a) {
    v2f A; A[0] = 1.0f; A[1] = 1.0f;
    v2f B; B[0] = a;    B[1] = 0.0f;
    v8f C = {};
    C = __builtin_amdgcn_wmma_f32_16x16x4_f32(false, A, false, B,
                                              (short)0, C, false, false);
    float r = C[0];
    r = dpp_xadd<0xB1>(r);     // xor 1
    r = dpp_xadd<0x4E>(r);     // xor 2
    r = dpp_xadd<0x164>(r);    // xor 4
    r = dpp_xadd<0x168>(r);    // xor 8
    return r;
}

// block of 256 threads = 8 waves; result valid in thread 0
__device__ __forceinline__ float block_reduce(float v, float* smem) {
    float w = wave_reduce(v);
    int lane = threadIdx.x & 31;
    int wid  = threadIdx.x >> 5;
    __syncthreads();                 // protect smem reuse across calls
    if (lane == 0) smem[wid] = w;
    __syncthreads();
    float r = 0.0f;
    if (wid == 0) {
        r = (lane < 8) ? smem[lane] : 0.0f;
        r = dpp_xadd<0xB1>(r);
        r = dpp_xadd<0x4E>(r);
        r = dpp_xadd<0x164>(r);      // lane 0 holds block sum
    }
    return r;
}

// PIL 'L' conversion, exact integer emulation
__device__ __forceinline__ int gray8(float r, float g, float b) {
    int ir = (int)(r * 255.0f);
    int ig = (int)(g * 255.0f);
    int ib = (int)(b * 255.0f);
    return (ir * 299 + ig * 587 + ib * 114) / 1000;
}

// ---------------- Kernel A: gray sums + gray u8 cache + fused L2 ----------------
__global__ void kGrayL2(const float* __restrict__ o, const float* __restrict__ t,
                        float* __restrict__ gsO, float* __restrict__ gsT,
                        float* __restrict__ l2P,
                        unsigned int* __restrict__ grO, unsigned int* __restrict__ grT,
                        int useGray) {
    int img = blockIdx.x >> 3;
    int slc = blockIdx.x & 7;
    const float* ob = o + (size_t)img * IMGSTRIDE;
    const float* tb = t + (size_t)img * IMGSTRIDE;
    int base4 = slc * 2048;              // float4 index within plane [0,16384)
    float sumO = 0.0f, sumT = 0.0f, l2 = 0.0f;

    for (int it = 0; it < 8; ++it) {
        int p4 = base4 + it * TPB + threadIdx.x;
        size_t bo = (size_t)p4 * 4;
        float4 oR = *(const float4*)(ob + bo);
        float4 oG = *(const float4*)(ob + PLANE + bo);
        float4 oB = *(const float4*)(ob + 2 * PLANE + bo);
        float4 tR = *(const float4*)(tb + bo);
        float4 tG = *(const float4*)(tb + PLANE + bo);
        float4 tB = *(const float4*)(tb + 2 * PLANE + bo);
        float orr[4] = {oR.x, oR.y, oR.z, oR.w};
        float ogg[4] = {oG.x, oG.y, oG.z, oG.w};
        float obb[4] = {oB.x, oB.y, oB.z, oB.w};
        float trr[4] = {tR.x, tR.y, tR.z, tR.w};
        float tgg[4] = {tG.x, tG.y, tG.z, tG.w};
        float tbb[4] = {tB.x, tB.y, tB.z, tB.w};
        unsigned int pO = 0, pT = 0;
#pragma unroll
        for (int k = 0; k < 4; ++k) {
            int go = gray8(orr[k], ogg[k], obb[k]);
            int gt = gray8(trr[k], tgg[k], tbb[k]);
            sumO += (float)go;
            sumT += (float)gt;
            float d0 = orr[k] - trr[k];
            float d1 = ogg[k] - tgg[k];
            float d2 = obb[k] - tbb[k];
            l2 += d0 * d0 + d1 * d1 + d2 * d2;
            pO |= (unsigned int)go << (8 * k);
            pT |= (unsigned int)gt << (8 * k);
        }
        if (useGray) {
            grO[(size_t)img * 16384 + p4] = pO;
            grT[(size_t)img * 16384 + p4] = pT;
        }
    }

    __shared__ float smem[8];
    float rO = block_reduce(sumO, smem);
    float rT = block_reduce(sumT, smem);
    float rL = block_reduce(l2, smem);
    if (threadIdx.x == 0) {
        gsO[blockIdx.x] = rO;
        gsT[blockIdx.x] = rT;
        l2P[blockIdx.x] = rL;
    }
}

// ---------------- Kernel T: fold 8 partials/image into thresholds ----------------
__global__ void kThres(const float* __restrict__ gsO, const float* __restrict__ gsT,
                       float* __restrict__ thres) {
    int t = threadIdx.x;                 // 0..255
    const float* src = (t < IMGS) ? gsO : gsT;
    int img = t & (IMGS - 1);
    float s = 0.0f;
    for (int k = 0; k < 8; ++k) s += src[img * 8 + k];
    thres[t] = s * (1.0f / (float)PLANE);
}

// ---------------- Kernel B: gradient-map L1 ----------------
__device__ __forceinline__ float mval(unsigned int b, float th) {
    return ((float)b > th) ? 0.0f : 1.0f;
}

__device__ __forceinline__ float maskr(const float* base, int i, int j, float th) {
    if ((unsigned)i >= (unsigned)HH || (unsigned)j >= (unsigned)WW) return 0.0f;
    int off = (i << 9) + j;
    int g = gray8(base[off], base[PLANE + off], base[2 * PLANE + off]);
    return ((float)g > th) ? 0.0f : 1.0f;
}

__device__ __forceinline__ float glut(int n) {
    // sqrt(1e-6), sqrt(0.25+1e-6), sqrt(0.5+1e-6)
    const float L0  = 0.00099999998f;
    const float L1  = 0.50000101f;
    const float L2v = 0.70710748f;
    return (n == 0) ? L0 : ((n == 1) ? L1 : L2v);
}

__global__ void kGrad(const unsigned char* __restrict__ grO,
                      const unsigned char* __restrict__ grT,
                      const float* __restrict__ o, const float* __restrict__ t,
                      const float* __restrict__ thres, float* __restrict__ lgP,
                      int useGray) {
    int tid0 = blockIdx.x * TPB + threadIdx.x;
    float acc = 0.0f;

    if (useGray) {
        // vectorized: 4 horizontally-consecutive pixels per iteration
        for (int g = tid0; g < NGRP; g += NBB * TPB) {
            int img = g >> 14;            // 16384 groups per image
            int pg  = g & 16383;
            int i   = pg >> 7;            // row (128 u32 groups per row)
            int jg  = pg & 127;
            float thO = thres[img];
            float thT = thres[IMGS + img];
            const unsigned char* gOb = grO + ((size_t)img << 16) + (i << 9) + (jg << 2);
            const unsigned char* gTb = grT + ((size_t)img << 16) + (i << 9) + (jg << 2);
            bool hasUp = (i > 0), hasDn = (i < HH - 1);
            unsigned int cO = *(const unsigned int*)gOb;
            unsigned int cT = *(const unsigned int*)gTb;
            unsigned int uO = hasUp ? *(const unsigned int*)(gOb - WW) : 0u;
            unsigned int uT = hasUp ? *(const unsigned int*)(gTb - WW) : 0u;
            unsigned int dO = hasDn ? *(const unsigned int*)(gOb + WW) : 0u;
            unsigned int dT = hasDn ? *(const unsigned int*)(gTb + WW) : 0u;
            float mLO = (jg > 0)   ? mval(gOb[-1], thO) : 0.0f;
            float mLT = (jg > 0)   ? mval(gTb[-1], thT) : 0.0f;
            float mRO = (jg < 127) ? mval(gOb[4], thO)  : 0.0f;
            float mRT = (jg < 127) ? mval(gTb[4], thT)  : 0.0f;

            float mO[4], mT[4];
#pragma unroll
            for (int k = 0; k < 4; ++k) {
                mO[k] = mval((cO >> (8 * k)) & 255u, thO);
                mT[k] = mval((cT >> (8 * k)) & 255u, thT);
            }
            float lO[4] = {mLO, mO[0], mO[1], mO[2]};
            float rO_[4] = {mO[1], mO[2], mO[3], mRO};
            float lT[4] = {mLT, mT[0], mT[1], mT[2]};
            float rT_[4] = {mT[1], mT[2], mT[3], mRT};
#pragma unroll
            for (int k = 0; k < 4; ++k) {
                float tO = hasUp ? mval((uO >> (8 * k)) & 255u, thO) : 0.0f;
                float bO = hasDn ? mval((dO >> (8 * k)) & 255u, thO) : 0.0f;
                float tT = hasUp ? mval((uT >> (8 * k)) & 255u, thT) : 0.0f;
                float bT = hasDn ? mval((dT >> (8 * k)) & 255u, thT) : 0.0f;
                int no = (rO_[k] != lO[k]) + (tO != bO);
                int nt = (rT_[k] != lT[k]) + (tT != bT);
                acc += fabsf(glut(no) - glut(nt));
            }
        }
    } else {
        // fallback: recompute grayscale from images (ws too small for cache)
        for (int idx = tid0; idx < NPIX; idx += NBB * TPB) {
            int img = idx >> 16;
            int p = idx & 65535;
            int i = p >> 9;
            int j = p & 511;
            float thO = thres[img];
            float thT = thres[IMGS + img];
            const float* iO = o + (size_t)img * IMGSTRIDE;
            const float* iT = t + (size_t)img * IMGSTRIDE;
            float ro = maskr(iO, i, j + 1, thO), lo = maskr(iO, i, j - 1, thO);
            float to = maskr(iO, i - 1, j, thO), bo = maskr(iO, i + 1, j, thO);
            float rt = maskr(iT, i, j + 1, thT), lt = maskr(iT, i, j - 1, thT);
            float tt = maskr(iT, i - 1, j, thT), bt = maskr(iT, i + 1, j, thT);
            int no = (ro != lo) + (to != bo);
            int nt = (rt != lt) + (tt != bt);
            acc += fabsf(glut(no) - glut(nt));
        }
    }

    __shared__ float smem[8];
    float r = block_reduce(acc, smem);
    if (threadIdx.x == 0) lgP[blockIdx.x] = r;
}

// ---------------- Kernel C: one-pass online log-softmax CE (one row / block) ----------------
__global__ void kCE(const float* __restrict__ pred, const int* __restrict__ gt,
                    float* __restrict__ ceR) {
    int row = blockIdx.x;
    const float* x = pred + (size_t)row * COLS;
    float xg = 0.0f;
    if (threadIdx.x == 0) xg = x[gt[row]];

    const float4* x4 = (const float4*)x;
    // 4 independent online-softmax accumulators to break the exp dep chain
    float m[4] = {-INFINITY, -INFINITY, -INFINITY, -INFINITY};
    float s[4] = {0.0f, 0.0f, 0.0f, 0.0f};
    for (int c = threadIdx.x; c < COLS4; c += TPB) {
        float4 v4 = x4[c];
        float v[4] = {v4.x, v4.y, v4.z, v4.w};
#pragma unroll
        for (int k = 0; k < 4; ++k) {
            float nm = fmaxf(m[k], v[k]);
            s[k] = s[k] * __expf(m[k] - nm) + __expf(v[k] - nm);
            m[k] = nm;
        }
    }
    // merge the 4 lanes
    float M = m[0], S = s[0];
#pragma unroll
    for (int k = 1; k < 4; ++k) {
        float nm = fmaxf(M, m[k]);
        S = S * __expf(M - nm) + s[k] * __expf(m[k] - nm);
        M = nm;
    }
    // wave32 (m,s) merge
#pragma unroll
    for (int off = 16; off > 0; off >>= 1) {
        float mo = __shfl_xor(M, off, 32);
        float so = __shfl_xor(S, off, 32);
        float nm = fmaxf(M, mo);
        S = S * __expf(M - nm) + so * __expf(mo - nm);
        M = nm;
    }
    __shared__ float sm[8], ss[8];
    int lane = threadIdx.x & 31, wid = threadIdx.x >> 5;
    if (lane == 0) { sm[wid] = M; ss[wid] = S; }
    __syncthreads();
    if (threadIdx.x == 0) {
        float MM = sm[0], SS = ss[0];
        for (int w = 1; w < 8; ++w) {
            float nm = fmaxf(MM, sm[w]);
            SS = SS * __expf(MM - nm) + ss[w] * __expf(sm[w] - nm);
            MM = nm;
        }
        ceR[row] = xg - MM - __logf(SS);   // log p(gt)
    }
}

// ---------------- Kernel F: deterministic final combine ----------------
__global__ void kFinal(const float* __restrict__ l2P, const float* __restrict__ lgP,
                       const float* __restrict__ ceR, float* __restrict__ out) {
    float a = 0.0f, b = 0.0f, c = 0.0f;
    for (int i = threadIdx.x; i < NBA; i += TPB) a += l2P[i];
    for (int i = threadIdx.x; i < NBB; i += TPB) b += lgP[i];
    for (int i = threadIdx.x; i < ROWS; i += TPB) c += ceR[i];
    __shared__ float smem[8];
    float sa = block_reduce(a, smem);
    float sb = block_reduce(b, smem);
    float sc = block_reduce(c, smem);
    if (threadIdx.x == 0) {
        float l2 = sa * (1.0f / 25165824.0f);       // B*3*H*W
        float lg = sb * (1.0f / 8388608.0f);        // 3-channel broadcast => 1-channel mean
        float ce = -(sc * (1.0f / 4096.0f));
        out[0] = 1.0f * l2 + 1e-4f * lg + 1e-5f * ce;
    }
}

extern "C" void kernel_launch(void* const* d_in, const int* in_sizes, int n_in,
                              void* d_out, int out_size, void* d_ws, size_t ws_size,
                              hipStream_t stream) {
    const float* outImg = (const float*)d_in[0];
    const float* tgtImg = (const float*)d_in[1];
    const float* pred   = (const float*)d_in[2];
    const int*   gt     = (const int*)d_in[3];

    float* ws    = (float*)d_ws;
    float* thres = ws + OFF_THRES;
    float* gsO   = ws + OFF_GS_O;
    float* gsT   = ws + OFF_GS_T;
    float* l2P   = ws + OFF_L2;
    float* lgP   = ws + OFF_LG;
    float* ceR   = ws + OFF_CE;

    int useGray = (ws_size >= (size_t)WS_NEED) ? 1 : 0;
    unsigned int* grO = nullptr;
    unsigned int* grT = nullptr;
    if (useGray) {
        grO = (unsigned int*)((unsigned char*)d_ws + GRAY_BYTE_OFF);
        grT = (unsigned int*)((unsigned char*)d_ws + GRAY_BYTE_OFF + NPIX);
    }

    kGrayL2<<<NBA, TPB, 0, stream>>>(outImg, tgtImg, gsO, gsT, l2P, grO, grT, useGray);
    kThres<<<1, TPB, 0, stream>>>(gsO, gsT, thres);
    kGrad<<<NBB, TPB, 0, stream>>>((const unsigned char*)grO, (const unsigned char*)grT,
                                   outImg, tgtImg, thres, lgP, useGray);
    kCE<<<ROWS, TPB, 0, stream>>>(pred, gt, ceR);
    kFinal<<<1, TPB, 0, stream>>>(l2P, lgP, ceR, (float*)d_out);
}